// Apriel2Mamba_40278203302192
// MI455X (gfx1250) — compile-verified
//
#include <hip/hip_runtime.h>
#include <math.h>

// ---------------------------------------------------------------------------
// Apriel2 Mamba layer for MI455X (gfx1250, wave32, WGP).
//  - GEMMs: LDS-tiled fp32 WMMA (v_wmma_f32_16x16x4_f32), 128x128 block tile,
//    8 waves/block, 2x4 register tiles per wave, BK=32 LDS chunks.
//  - LDS staging via global_load_async_to_lds_b128 (ASYNCcnt) when available.
//  - Selective scan: one wave32 per (batch, head); 16x16 state in 8 VGPRs/lane.
// ---------------------------------------------------------------------------

#define D_MODEL  2048
#define D_INNER  4096
#define D_XB     1024
#define D_STATE  16
#define D_CONV   4
#define DT_RANK  128
#define BATCH    2
#define SEQLEN   2048
#define ZXBC_W   (2*D_XB + 2*D_INNER) // 10240
#define M_ROWS   (BATCH*SEQLEN)       // 4096

// zxbc column layout: z [0,4096) | x [4096,5120) | B [5120,6144) | C [6144,10240)
#define OFF_Z 0
#define OFF_X D_INNER
#define OFF_B (D_INNER + D_XB)
#define OFF_C (D_INNER + 2*D_XB)

typedef __attribute__((ext_vector_type(2))) float v2f;
typedef __attribute__((ext_vector_type(8))) float v8f;
typedef int v4i_b128 __attribute__((vector_size(16)));   // matches builtin param

// 32 data cols + 4 pad: rows stay 16B-aligned (36*4=144 = 9*16) and lane
// fragment reads (row stride 36 mod 64 banks) hit 16 distinct banks.
#define LDS_STRIDE 36

#if __has_builtin(__builtin_amdgcn_global_load_async_to_lds_b128)
#define ASYNC_LDS 1
#else
#define ASYNC_LDS 0
#endif

__device__ __forceinline__ void async_wait_all() {
#if ASYNC_LDS
#if __has_builtin(__builtin_amdgcn_s_wait_asynccnt)
  __builtin_amdgcn_s_wait_asynccnt(0);
#else
  asm volatile("s_wait_asynccnt 0x0" ::: "memory");
#endif
#endif
}

__device__ __forceinline__ void stage_to_lds(const float* __restrict__ g,
                                             float* __restrict__ l) {
#if ASYNC_LDS
  __builtin_amdgcn_global_load_async_to_lds_b128(
      (__attribute__((address_space(1))) v4i_b128*)(v4i_b128*)(float*)g,
      (__attribute__((address_space(3))) v4i_b128*)(v4i_b128*)l, 0, 0);
#else
  *(float4*)l = *(const float4*)g;
#endif
}

// ---------------------------------------------------------------------------
// LDS-tiled WMMA GEMM:  Cout(MxN) = A(MxK) @ B(NxK)^T   (both row-major, K-major)
// Block: 128x128 tile, 8 waves; wave (wr,wc) owns 2 M-tiles x 4 N-tiles.
// EPI==1: out = softplus(acc + bias[n]) fused epilogue (dt path).
// ---------------------------------------------------------------------------
template <int EPI>
__global__ __launch_bounds__(256)
void k_gemm_wmma(const float* __restrict__ A, const float* __restrict__ B,
                 const float* __restrict__ bias, float* __restrict__ Cout,
                 int lda, int ldb, int ldo, int K, int tilesN) {
  __shared__ float sA[128 * LDS_STRIDE];
  __shared__ float sB[128 * LDS_STRIDE];

  const int tid  = threadIdx.x;
  const int lane = tid & 31;
  const int w    = tid >> 5;       // 0..7
  const int wr   = w >> 1;         // 0..3 : 32-row band
  const int wc   = w & 1;          // 0..1 : 64-col half
  const int r    = lane & 15;
  const int hi   = lane >> 4;      // K sub-pair select per 16x4 f32 layout

  const int m0 = (blockIdx.x / tilesN) * 128;
  const int n0 = (blockIdx.x % tilesN) * 128;

  // cooperative loader: each thread moves 4 float4 rows per operand
  const int lrow = tid >> 3;       // 0..31
  const int lcol = (tid & 7) * 4;  // 0..28

  v8f c[2][4];
#pragma unroll
  for (int m = 0; m < 2; ++m)
#pragma unroll
    for (int n = 0; n < 4; ++n) c[m][n] = (v8f){};

  for (int k0 = 0; k0 < K; k0 += 32) {
#pragma unroll
    for (int i = 0; i < 4; ++i) {
      const int row = lrow + i * 32;
      stage_to_lds(A + (size_t)(m0 + row) * lda + k0 + lcol,
                   &sA[row * LDS_STRIDE + lcol]);
      stage_to_lds(B + (size_t)(n0 + row) * ldb + k0 + lcol,
                   &sB[row * LDS_STRIDE + lcol]);
    }
    async_wait_all();
    __syncthreads();

#pragma unroll
    for (int k4 = 0; k4 < 32; k4 += 4) {
      v2f a[2], b[4];
#pragma unroll
      for (int m = 0; m < 2; ++m)
        a[m] = *(const v2f*)&sA[(wr * 32 + m * 16 + r) * LDS_STRIDE + k4 + 2 * hi];
#pragma unroll
      for (int n = 0; n < 4; ++n)
        b[n] = *(const v2f*)&sB[(wc * 64 + n * 16 + r) * LDS_STRIDE + k4 + 2 * hi];
#pragma unroll
      for (int m = 0; m < 2; ++m)
#pragma unroll
        for (int n = 0; n < 4; ++n)
          c[m][n] = __builtin_amdgcn_wmma_f32_16x16x4_f32(
              false, a[m], false, b[n], (short)0, c[m][n], false, false);
    }
    __syncthreads();
  }

#pragma unroll
  for (int m = 0; m < 2; ++m)
#pragma unroll
    for (int n = 0; n < 4; ++n) {
      const int gn = n0 + wc * 64 + n * 16 + r;
      const float bb = (EPI == 1) ? bias[gn] : 0.0f;
      float* outp = Cout + (size_t)(m0 + wr * 32 + m * 16 + 8 * hi) * ldo + gn;
#pragma unroll
      for (int v = 0; v < 8; ++v) {
        float x = c[m][n][v];
        if (EPI == 1) {
          x += bb;
          x = (x > 20.0f) ? x : log1pf(__expf(x));   // softplus
        }
        outp[(size_t)v * ldo] = x;
      }
    }
}

// ---------------------------------------------------------------------------
// GQA-repeat + causal depthwise conv1d(k=4) + SiLU  -> u (b,l,d_inner)
// source x channel of d_inner channel d: ((d>>6)<<4) | (d&15)
// ---------------------------------------------------------------------------
__global__ void k_conv_silu(const float* __restrict__ zxbc,
                            const float* __restrict__ cw,
                            const float* __restrict__ cb,
                            float* __restrict__ u) {
  const size_t idx = (size_t)blockIdx.x * blockDim.x + threadIdx.x;
  const int d = (int)(idx % D_INNER);
  const int t = (int)((idx / D_INNER) % SEQLEN);
  const int b = (int)(idx / ((size_t)D_INNER * SEQLEN));
  const int xc = ((d >> 6) << 4) | (d & 15);
  const float* base = zxbc + (size_t)b * SEQLEN * ZXBC_W + OFF_X + xc;
  float acc = cb[d];
#pragma unroll
  for (int k = 0; k < D_CONV; ++k) {
    const int tt = t - (D_CONV - 1) + k;
    if (tt >= 0) acc = fmaf(cw[d * D_CONV + k], base[(size_t)tt * ZXBC_W], acc);
  }
  u[idx] = acc / (1.0f + __expf(-acc));   // SiLU
}

// ---------------------------------------------------------------------------
// Selective scan + skip + gate.  One wave32 per (b, head).
// lane = n_hi*16 + p ; lane keeps s[p, n_hi*8+j], j=0..7 (8 VGPRs).
// y_p = sum_n s*C  -> shfl_xor(16) cross-half reduce, fused (y+u*D)*silu(z).
// ---------------------------------------------------------------------------
__global__ void k_scan(const float* __restrict__ zxbc,
                       const float* __restrict__ u,
                       const float* __restrict__ dt,
                       const float* __restrict__ A_log,
                       const float* __restrict__ Dskip,
                       float* __restrict__ y) {
  const int lane = threadIdx.x & 31;
  const int wid  = blockIdx.x * (blockDim.x >> 5) + (threadIdx.x >> 5); // 0..511
  const int b    = wid >> 8;
  const int head = wid & 255;
  const int p    = lane & 15;
  const int hi   = lane >> 4;
  const int d    = head * 16 + p;

  float Aj[8];
#pragma unroll
  for (int j = 0; j < 8; ++j)
    Aj[j] = -__expf(A_log[(size_t)d * D_STATE + hi * 8 + j]);
  const float Dv = Dskip[d];

  float s[8];
#pragma unroll
  for (int j = 0; j < 8; ++j) s[j] = 0.0f;

  const size_t rowZ = (size_t)b * SEQLEN * ZXBC_W;
  const size_t rowU = (size_t)b * SEQLEN * D_INNER;
  const float* Bbase  = zxbc + rowZ + OFF_B + (head >> 2) * 16 + hi * 8;
  const float* Cbase  = zxbc + rowZ + OFF_C + head * 16 + hi * 8;
  const float* zbase  = zxbc + rowZ + OFF_Z + d;
  const float* ubase  = u  + rowU + d;
  const float* dtbase = dt + rowU + d;
  float*       ybase  = y  + rowU + d;

  for (int t = 0; t < SEQLEN; ++t) {
    const size_t rz = (size_t)t * ZXBC_W;
    const size_t ru = (size_t)t * D_INNER;
    const float dtp = dtbase[ru];
    const float up  = ubase[ru];
    const float4 B0 = *(const float4*)(Bbase + rz);
    const float4 B1 = *(const float4*)(Bbase + rz + 4);
    const float4 C0 = *(const float4*)(Cbase + rz);
    const float4 C1 = *(const float4*)(Cbase + rz + 4);
    const float Bv[8] = {B0.x, B0.y, B0.z, B0.w, B1.x, B1.y, B1.z, B1.w};
    const float Cv[8] = {C0.x, C0.y, C0.z, C0.w, C1.x, C1.y, C1.z, C1.w};
    const float dtu = dtp * up;
    float acc = 0.0f;
#pragma unroll
    for (int j = 0; j < 8; ++j) {
      const float dA = __expf(dtp * Aj[j]);
      s[j] = fmaf(dA, s[j], dtu * Bv[j]);
      acc  = fmaf(s[j], Cv[j], acc);
    }
    acc += __shfl_xor(acc, 16, 32);       // reduce across n halves
    if (hi == 0) {
      const float zv = zbase[rz];
      const float g  = zv / (1.0f + __expf(-zv));   // silu(z)
      ybase[ru] = (acc + up * Dv) * g;
    }
  }
}

// ---------------------------------------------------------------------------
// Workspace layout (floats)
// ---------------------------------------------------------------------------
#define WS_ZXBC   ((size_t)0)
#define WS_U      (WS_ZXBC + (size_t)M_ROWS * ZXBC_W)
#define WS_DTTMP  (WS_U + (size_t)M_ROWS * D_INNER)
#define WS_DT     (WS_DTTMP + (size_t)M_ROWS * DT_RANK)
#define WS_Y      (WS_DT + (size_t)M_ROWS * D_INNER)

extern "C" void kernel_launch(void* const* d_in, const int* in_sizes, int n_in,
                              void* d_out, int out_size, void* d_ws, size_t ws_size,
                              hipStream_t stream) {
  const float* hs        = (const float*)d_in[0];
  const float* in_proj_w = (const float*)d_in[1];
  const float* dt_in_w   = (const float*)d_in[2];
  const float* dt_proj_w = (const float*)d_in[3];
  const float* dt_proj_b = (const float*)d_in[4];
  const float* conv_w    = (const float*)d_in[5];
  const float* conv_b    = (const float*)d_in[6];
  const float* A_log     = (const float*)d_in[7];
  const float* Dskip     = (const float*)d_in[8];
  const float* out_w     = (const float*)d_in[9];
  float* out = (float*)d_out;

  float* ws     = (float*)d_ws;
  float* zxbc   = ws + WS_ZXBC;
  float* u      = ws + WS_U;
  float* dt_tmp = ws + WS_DTTMP;
  float* dtb    = ws + WS_DT;
  float* y      = ws + WS_Y;

  const int THREADS = 256;
  const int TM = M_ROWS / 128;          // 32 block rows

  // 1) zxbc = hidden @ in_proj_w.T           (4096 x 10240, K=2048)
  k_gemm_wmma<0><<<TM * (ZXBC_W / 128), THREADS, 0, stream>>>(
      hs, in_proj_w, nullptr, zxbc, D_MODEL, D_MODEL, ZXBC_W, D_MODEL, ZXBC_W / 128);
  // 2a) dt_tmp = hidden @ dt_in_proj_w.T     (4096 x 128, K=2048)
  k_gemm_wmma<0><<<TM * (DT_RANK / 128), THREADS, 0, stream>>>(
      hs, dt_in_w, nullptr, dt_tmp, D_MODEL, D_MODEL, DT_RANK, D_MODEL, DT_RANK / 128);
  // 2b) dt = softplus(dt_tmp @ dt_proj_w.T + b)  (4096 x 4096, K=128)
  k_gemm_wmma<1><<<TM * (D_INNER / 128), THREADS, 0, stream>>>(
      dt_tmp, dt_proj_w, dt_proj_b, dtb, DT_RANK, DT_RANK, D_INNER, DT_RANK, D_INNER / 128);
  // 3) repeat + conv + silu
  k_conv_silu<<<((size_t)BATCH * SEQLEN * D_INNER) / THREADS, THREADS, 0, stream>>>(
      zxbc, conv_w, conv_b, u);
  // 4) selective scan (512 waves)
  k_scan<<<512 / (THREADS / 32), THREADS, 0, stream>>>(zxbc, u, dtb, A_log, Dskip, y);
  // 5) out = y @ out_proj_w.T                (4096 x 2048, K=4096)
  k_gemm_wmma<0><<<TM * (D_MODEL / 128), THREADS, 0, stream>>>(
      y, out_w, nullptr, out, D_INNER, D_INNER, D_MODEL, D_INNER, D_MODEL / 128);
}